// DenseGTVConv_18915035972353
// MI455X (gfx1250) — compile-verified
//
#include <hip/hip_runtime.h>
#include <hip/hip_bf16.h>

typedef __attribute__((ext_vector_type(2))) float v2f;
typedef __attribute__((ext_vector_type(8))) float v8f;
typedef unsigned int u32x4 __attribute__((ext_vector_type(4)));
typedef int i32x4 __attribute__((ext_vector_type(4)));
typedef int i32x8 __attribute__((ext_vector_type(8)));

#define CLAMP_MIN 1e-3f
#define KT 64            // K-tile per TDM transfer
#define APITCH 66        // LDS row pitch (floats): 64 + 2 pad -> 8B aligned, conflict-free

#ifndef __has_builtin
#define __has_builtin(x) 0
#endif
#if __has_builtin(__builtin_amdgcn_tensor_load_to_lds)
#define HAVE_TDM 1
#else
#define HAVE_TDM 0
#endif

// ---------------------------------------------------------------------------
// TDM: issue one 2D tile load mod[16 rows x KT cols] -> LDS, row pitch APITCH.
// D# per CDNA5 ISA 8.3/8.4:
//   group0: count=1 | lds_addr | global_addr(57b) | type=2
//   group1: data_size=4B, pad_enable, pad_interval=5 (64 DW), pad_amount=1 (2 DW),
//           tensor_dim0=tensor_dim1=N, tile_dim0=KT, tile_dim1=16,
//           tensor_dim0_stride=N
// ---------------------------------------------------------------------------
#if HAVE_TDM
__device__ __forceinline__ void tdm_load_tile(unsigned lds_off,
                                              const float* gptr, int N) {
  unsigned long long ga = (unsigned long long)(size_t)gptr;
  u32x4 g0;
  g0[0] = 1u;                                            // count=1 (user D#)
  g0[1] = lds_off;                                       // LDS byte address
  g0[2] = (unsigned)ga;                                  // global_addr[31:0]
  g0[3] = (unsigned)((ga >> 32) & 0x01FFFFFFu) | (2u << 30); // [56:32] | type=2
  const unsigned td = (unsigned)N;
  i32x8 g1;
  g1[0] = (int)((2u << 16) | (1u << 20) | (5u << 22) | (1u << 25));
  //            data_size=4B  pad_en      pad_iv=64DW   pad_amt=2DW
  g1[1] = (int)((td & 0xFFFFu) << 16);                   // tensor_dim0[15:0]
  g1[2] = (int)(((td >> 16) & 0xFFFFu) | ((td & 0xFFFFu) << 16)); // td0_hi|td1_lo
  g1[3] = (int)(((td >> 16) & 0xFFFFu) | ((unsigned)KT << 16));   // td1_hi|tile_dim0
  g1[4] = 16;                                            // tile_dim1=16, tile_dim2=0
  g1[5] = (int)(unsigned)N;                              // tensor_dim0_stride lo
  g1[6] = 0;                                             // stride hi | dim1_stride lo
  g1[7] = 0;                                             // dim1_stride hi
  i32x4 z4 = (i32x4)0;
#if defined(__clang_major__) && (__clang_major__ >= 23)
  i32x8 z8 = (i32x8)0;
  __builtin_amdgcn_tensor_load_to_lds(g0, g1, z4, z4, z8, 0);
#else
  __builtin_amdgcn_tensor_load_to_lds(g0, g1, z4, z4, 0);
#endif
}
#endif

// ---------------------------------------------------------------------------
// Kernel 1: y = x @ W  — one wave per 16x16 tile, V_WMMA_F32_16X16X4_F32.
// A layout (16x4 f32): lanes 0-15 -> K={0,1}; lanes 16-31 -> K={2,3} (b64/lane).
// ---------------------------------------------------------------------------
__global__ void __launch_bounds__(128) k_gemm_xw(const float* __restrict__ x,
                                                 const float* __restrict__ w,
                                                 float* __restrict__ y,
                                                 int N, int Fin, int Fout) {
  const int b    = blockIdx.y;
  const int i0   = blockIdx.x * 16;
  const int wv   = threadIdx.x >> 5;
  const int lane = threadIdx.x & 31;
  const int f0   = wv * 16;
  const int row  = lane & 15;          // also B/D column (lane % 16)
  const int kh   = (lane >> 4) << 1;   // 0 or 2

  const float* xrow = x + ((size_t)b * N + i0 + row) * Fin;
  v8f c = {};
  for (int k0 = 0; k0 < Fin; k0 += 4) {
    v2f a = *(const v2f*)(xrow + k0 + kh);
    const float* wk = w + (size_t)(k0 + kh) * Fout + f0 + row;
    v2f bb;
    bb.x = wk[0];
    bb.y = wk[Fout];
    c = __builtin_amdgcn_wmma_f32_16x16x4_f32(false, a, false, bb,
                                              (short)0, c, false, false);
  }
  const int mh = (lane >> 4) << 3;
  float* orow = y + ((size_t)b * N + i0 + mh) * Fout + f0 + row;
#pragma unroll
  for (int r = 0; r < 8; ++r) orow[(size_t)r * Fout] = c[r];
}

// ---------------------------------------------------------------------------
// Kernel 2: mod = I - diag(deg) + adj / max(L1dist(y_i,y_j), 1e-3)
// Block(256) owns 16 i-rows; loops all j. Deg + diagonal fix-up local.
// ---------------------------------------------------------------------------
__global__ void __launch_bounds__(256) k_mod(const float* __restrict__ y,
                                             const float* __restrict__ adj,
                                             float* __restrict__ mod,
                                             int N, int Fout) {
  __shared__ float yi[16][65];
  __shared__ float yj[16][65];
  __shared__ float dsum[16][17];

  const int b  = blockIdx.y;
  const int i0 = blockIdx.x * 16;
  const int t  = threadIdx.x;
  const int li = t >> 4;
  const int lj = t & 15;

  for (int idx = t; idx < 16 * 64; idx += 256)
    yi[idx >> 6][idx & 63] =
        y[((size_t)b * N + i0 + (idx >> 6)) * Fout + (idx & 63)];

  float degacc = 0.f;
  const size_t arow = ((size_t)b * N + i0 + li) * N;

  for (int j0 = 0; j0 < N; j0 += 16) {
    __syncthreads();
    for (int idx = t; idx < 16 * 64; idx += 256)
      yj[idx >> 6][idx & 63] =
          y[((size_t)b * N + j0 + (idx >> 6)) * Fout + (idx & 63)];
    __syncthreads();

    if (j0 + 16 < N)  // prefetch next adj chunk (8MB stream) -> global_prefetch
      __builtin_prefetch((const void*)(adj + arow + j0 + 16 + lj), 0, 0);

    float d = 0.f;
#pragma unroll
    for (int f = 0; f < 64; ++f) d += fabsf(yi[li][f] - yj[lj][f]);

    const float m = adj[arow + j0 + lj] / fmaxf(d, CLAMP_MIN);
    mod[arow + j0 + lj] = m;
    degacc += m;
  }

  __syncthreads();
  dsum[li][lj] = degacc;
  __syncthreads();
  if (t < 16) {
    float deg = 0.f;
#pragma unroll
    for (int q = 0; q < 16; ++q) deg += dsum[t][q];
    const size_t di = ((size_t)b * N + i0 + t) * N + (size_t)(i0 + t);
    mod[di] += 1.0f - deg;
  }
}

// ---------------------------------------------------------------------------
// Kernel 3: out = mod @ y + bias.
// A (mod) streamed by TDM into a double-buffered LDS tile [16][KT] with HW
// padding to pitch 66; wave 0 issues DMA + s_wait_tensorcnt; all 4 waves
// consume via ds b64 reads feeding V_WMMA_F32_16X16X4_F32.
// ---------------------------------------------------------------------------
__global__ void __launch_bounds__(128) k_gemm_out(const float* __restrict__ mod,
                                                  const float* __restrict__ y,
                                                  const float* __restrict__ bias,
                                                  float* __restrict__ out,
                                                  int N, int Fout) {
  __shared__ float tileA[2][16][APITCH];

  const int b    = blockIdx.y;
  const int i0   = blockIdx.x * 16;
  const int wv   = threadIdx.x >> 5;
  const int lane = threadIdx.x & 31;
  const int f0   = wv * 16;
  const int row  = lane & 15;
  const int kh   = (lane >> 4) << 1;

  const float* abase = mod + ((size_t)b * N + i0) * N;      // tile origin (K=0)
  const float* ybase = y + (size_t)b * N * Fout + f0 + row;

  v8f c = {};
  const int NT = N / KT;

#if HAVE_TDM
  if (wv == 0)
    tdm_load_tile((unsigned)(size_t)&tileA[0][0][0], abase, N);
  for (int it = 0; it < NT; ++it) {
    const int cur = it & 1;
    if (wv == 0) __builtin_amdgcn_s_wait_tensorcnt(0);
    __syncthreads();                                        // tile 'cur' ready
    if (wv == 0 && (it + 1) < NT)
      tdm_load_tile((unsigned)(size_t)&tileA[cur ^ 1][0][0],
                    abase + (size_t)(it + 1) * KT, N);      // overlap next DMA
    const float* ybk = ybase + (size_t)it * KT * Fout;
#pragma unroll
    for (int kk = 0; kk < KT; kk += 4) {
      v2f a = *(const v2f*)&tileA[cur][row][kk + kh];
      const float* yk = ybk + (size_t)(kk + kh) * Fout;
      v2f bb;
      bb.x = yk[0];
      bb.y = yk[Fout];
      c = __builtin_amdgcn_wmma_f32_16x16x4_f32(false, a, false, bb,
                                                (short)0, c, false, false);
    }
    __syncthreads();                                        // done reading 'cur'
  }
#else
  // Fallback: cooperative LDS fill (no TDM builtin on this toolchain)
  for (int it = 0; it < NT; ++it) {
    __syncthreads();
    for (int idx = threadIdx.x; idx < 16 * KT; idx += 128) {
      const int r = idx >> 6, k = idx & (KT - 1);
      tileA[0][r][k] = abase[(size_t)r * N + (size_t)it * KT + k];
    }
    __syncthreads();
    const float* ybk = ybase + (size_t)it * KT * Fout;
#pragma unroll
    for (int kk = 0; kk < KT; kk += 4) {
      v2f a = *(const v2f*)&tileA[0][row][kk + kh];
      const float* yk = ybk + (size_t)(kk + kh) * Fout;
      v2f bb;
      bb.x = yk[0];
      bb.y = yk[Fout];
      c = __builtin_amdgcn_wmma_f32_16x16x4_f32(false, a, false, bb,
                                                (short)0, c, false, false);
    }
  }
#endif

  const int mh = (lane >> 4) << 3;
  const float bv = bias[f0 + row];
  float* orow = out + ((size_t)b * N + i0 + mh) * Fout + f0 + row;
#pragma unroll
  for (int r = 0; r < 8; ++r) orow[(size_t)r * Fout] = c[r] + bv;
}

// ---------------------------------------------------------------------------
extern "C" void kernel_launch(void* const* d_in, const int* in_sizes, int n_in,
                              void* d_out, int out_size, void* d_ws, size_t ws_size,
                              hipStream_t stream) {
  const float* x    = (const float*)d_in[0];  // [B,N,Fin]
  const float* adj  = (const float*)d_in[1];  // [B,N,N]
  const float* w    = (const float*)d_in[2];  // [Fin,Fout]
  const float* bias = (const float*)d_in[3];  // [Fout]
  float* out = (float*)d_out;                 // [B,N,Fout]

  const int  Fout = in_sizes[3];                    // 64
  const int  Fin  = in_sizes[2] / Fout;             // 128
  const long BN   = (long)in_sizes[0] / Fin;        // B*N
  const int  N    = (int)((long)in_sizes[1] / BN);  // 1024
  const int  B    = (int)(BN / N);                  // 2

  float* y   = (float*)d_ws;                        // [B,N,Fout]
  float* mod = y + (size_t)B * N * Fout;            // [B,N,N]

  dim3 grid(N / 16, B);
  k_gemm_xw <<<grid, 128, 0, stream>>>(x, w, y, N, Fin, Fout);
  k_mod     <<<grid, 256, 0, stream>>>(y, adj, mod, N, Fout);
  k_gemm_out<<<grid, 128, 0, stream>>>(mod, y, bias, out, N, Fout);
}